// TX_18236431139551
// MI455X (gfx1250) — compile-verified
//
#include <hip/hip_runtime.h>
#include <math.h>

typedef float v2f __attribute__((ext_vector_type(2)));
typedef float v8f __attribute__((ext_vector_type(8)));

#define D_MODEL 64
#define D_FF    32
#define WAVES   8
#define TILE    16

// One block per batch row. 8 waves stream disjoint t-chunks with online
// softmax; scores computed with chained V_WMMA_F32_16X16X4_F32 (fp32-exact).
__global__ __launch_bounds__(256, 1) void fused_attn_ffn(
    const float* __restrict__ q,  const float* __restrict__ k,
    const float* __restrict__ v,
    const float* __restrict__ w1, const float* __restrict__ b1,
    const float* __restrict__ w2, const float* __restrict__ b2,
    const float* __restrict__ alpha1, const float* __restrict__ bias1,
    const float* __restrict__ alpha2, const float* __restrict__ bias2,
    float* __restrict__ out, int T)
{
    __shared__ float qs[D_MODEL];
    __shared__ float wm[WAVES], wl[WAVES];
    __shared__ float wacc[WAVES][D_MODEL];
    __shared__ float qn[D_MODEL];
    __shared__ float hs[D_FF];

    const int brow = blockIdx.x;
    const int tid  = threadIdx.x;
    const int lane = tid & 31;
    const int wave = tid >> 5;

    if (tid < D_MODEL) qs[tid] = q[brow * D_MODEL + tid];
    __syncthreads();

    const int half = lane >> 4;   // which K-pair this lane supplies (A & B layout)
    const int mrow = lane & 15;   // A-matrix row owned by this lane

    // B-matrix (4x16) registers: row K holds q[kbase+K] replicated across columns.
    // VGPR0 = {K0 | K2}, VGPR1 = {K1 | K3} across the two lane halves.
    v2f bq[16];
#pragma unroll
    for (int c = 0; c < 16; ++c) {
        bq[c][0] = qs[4 * c + 2 * half];
        bq[c][1] = qs[4 * c + 2 * half + 1];
    }

    float mrun = -3.3e38f, lrun = 0.f;
    float acc0 = 0.f, acc1 = 0.f;            // output dims 2*lane, 2*lane+1
    const size_t base = (size_t)brow * (size_t)T * D_MODEL;
    const int t_per_wave = T / WAVES;
    const int t_begin = wave * t_per_wave;

    for (int t0 = t_begin; t0 < t_begin + t_per_wave; t0 += TILE) {
        const float* kr = k + base + (size_t)(t0 + mrow) * D_MODEL + 2 * half;
        const float* vr = v + base + (size_t)t0 * D_MODEL + 2 * lane;
        __builtin_prefetch(kr + TILE * D_MODEL, 0, 1);   // global_prefetch_b8
        __builtin_prefetch(vr + TILE * D_MODEL, 0, 1);

        // scores for 16 t-rows: chain 16 WMMA f32 16x16x4 over K=64
        v8f c8 = {0.f, 0.f, 0.f, 0.f, 0.f, 0.f, 0.f, 0.f};
#pragma unroll
        for (int c = 0; c < 16; ++c) {
            v2f a = *(const v2f*)(kr + 4 * c);   // b64 load, every byte used
            c8 = __builtin_amdgcn_wmma_f32_16x16x4_f32(
                    false, a, false, bq[c], (short)0, c8, false, false);
        }

        // C layout: lane<16 holds rows 0..7 in c8[0..7]; lane>=16 rows 8..15.
        float s[8];
        float tmax = -3.3e38f;
#pragma unroll
        for (int i = 0; i < 8; ++i) { s[i] = c8[i] * 0.125f; tmax = fmaxf(tmax, s[i]); }
        tmax = fmaxf(tmax, __shfl_xor(tmax, 16, 32));
        const float mnew = fmaxf(mrun, tmax);
        const float corr = __expf(mrun - mnew);
        float p[8], psum = 0.f;
#pragma unroll
        for (int i = 0; i < 8; ++i) { p[i] = __expf(s[i] - mnew); psum += p[i]; }
        psum += __shfl_xor(psum, 16, 32);
        lrun = lrun * corr + psum;
        mrun = mnew;
        acc0 *= corr; acc1 *= corr;

        float po[8];
#pragma unroll
        for (int i = 0; i < 8; ++i) po[i] = __shfl_xor(p[i], 16, 32);
        const bool low = (half == 0);
#pragma unroll
        for (int i = 0; i < 8; ++i) {
            const float plo = low ? p[i]  : po[i];   // weight for row i
            const float phi = low ? po[i] : p[i];    // weight for row i+8
            v2f vlo = *(const v2f*)(vr + (size_t)i * D_MODEL);        // coalesced b64
            v2f vhi = *(const v2f*)(vr + (size_t)(i + 8) * D_MODEL);
            acc0 = fmaf(plo, vlo[0], acc0); acc1 = fmaf(plo, vlo[1], acc1);
            acc0 = fmaf(phi, vhi[0], acc0); acc1 = fmaf(phi, vhi[1], acc1);
        }
    }

    if (lane == 0) { wm[wave] = mrun; wl[wave] = lrun; }
    wacc[wave][2 * lane]     = acc0;
    wacc[wave][2 * lane + 1] = acc1;
    __syncthreads();

    // ---- wave 0: merge partials, softmax-normalize, add residual, norm1 ----
    if (wave == 0) {
        float M = wm[0];
#pragma unroll
        for (int w = 1; w < WAVES; ++w) M = fmaxf(M, wm[w]);
        float L = 0.f, A0 = 0.f, A1 = 0.f;
#pragma unroll
        for (int w = 0; w < WAVES; ++w) {
            const float e = __expf(wm[w] - M);
            L += e * wl[w];
            A0 = fmaf(e, wacc[w][2 * lane],     A0);
            A1 = fmaf(e, wacc[w][2 * lane + 1], A1);
        }
        const float rinv = 1.f / L;
        const float x0 = A0 * rinv + qs[2 * lane];
        const float x1 = A1 * rinv + qs[2 * lane + 1];
        float sm = x0 + x1, s2 = x0 * x0 + x1 * x1;
#pragma unroll
        for (int off = 16; off > 0; off >>= 1) {
            sm += __shfl_xor(sm, off, 32);
            s2 += __shfl_xor(s2, off, 32);
        }
        const float mu  = sm * (1.f / 64.f);
        const float var = (s2 - 64.f * mu * mu) * (1.f / 63.f);  // ddof=1
        const float inv = 1.f / (sqrtf(fmaxf(var, 0.f)) + 1e-6f);
        qn[2 * lane]     = alpha1[2 * lane]     * (x0 - mu) * inv + bias1[2 * lane];
        qn[2 * lane + 1] = alpha1[2 * lane + 1] * (x1 - mu) * inv + bias1[2 * lane + 1];
    }
    __syncthreads();

    // ---- FFN hidden: h[j] = relu(qn . w1[j,:] + b1[j]), one lane per j ----
    if (wave == 0) {
        float hv = b1[lane];
        const float* w1r = w1 + lane * D_MODEL;
#pragma unroll 8
        for (int d = 0; d < D_MODEL; ++d) hv = fmaf(qn[d], w1r[d], hv);
        hs[lane] = fmaxf(hv, 0.f);
    }
    __syncthreads();

    // ---- FFN out + residual + norm2 + store ----
    if (wave == 0) {
        float y0 = b2[2 * lane], y1 = b2[2 * lane + 1];
        const float* w2r0 = w2 + (size_t)(2 * lane) * D_FF;
        const float* w2r1 = w2 + (size_t)(2 * lane + 1) * D_FF;
#pragma unroll 8
        for (int j = 0; j < D_FF; ++j) {
            const float hj = hs[j];
            y0 = fmaf(hj, w2r0[j], y0);
            y1 = fmaf(hj, w2r1[j], y1);
        }
        const float x0 = qn[2 * lane] + y0;
        const float x1 = qn[2 * lane + 1] + y1;
        float sm = x0 + x1, s2 = x0 * x0 + x1 * x1;
#pragma unroll
        for (int off = 16; off > 0; off >>= 1) {
            sm += __shfl_xor(sm, off, 32);
            s2 += __shfl_xor(s2, off, 32);
        }
        const float mu  = sm * (1.f / 64.f);
        const float var = (s2 - 64.f * mu * mu) * (1.f / 63.f);
        const float inv = 1.f / (sqrtf(fmaxf(var, 0.f)) + 1e-6f);
        v2f r;
        r[0] = alpha2[2 * lane]     * (x0 - mu) * inv + bias2[2 * lane];
        r[1] = alpha2[2 * lane + 1] * (x1 - mu) * inv + bias2[2 * lane + 1];
        *(v2f*)(out + (size_t)brow * D_MODEL + 2 * lane) = r;
    }
}

extern "C" void kernel_launch(void* const* d_in, const int* in_sizes, int n_in,
                              void* d_out, int out_size, void* d_ws, size_t ws_size,
                              hipStream_t stream) {
    const float* q      = (const float*)d_in[0];
    const float* k      = (const float*)d_in[1];
    const float* v      = (const float*)d_in[2];
    const float* w1     = (const float*)d_in[3];
    const float* b1     = (const float*)d_in[4];
    const float* w2     = (const float*)d_in[5];
    const float* b2     = (const float*)d_in[6];
    const float* alpha1 = (const float*)d_in[7];
    const float* bias1  = (const float*)d_in[8];
    const float* alpha2 = (const float*)d_in[9];
    const float* bias2  = (const float*)d_in[10];
    float* out = (float*)d_out;

    const int B = in_sizes[0] / D_MODEL;        // batch rows
    const int T = in_sizes[1] / in_sizes[0];    // (b*t*d)/(b*d) = t

    fused_attn_ffn<<<dim3(B), dim3(256), 0, stream>>>(
        q, k, v, w1, b1, w2, b2, alpha1, bias1, alpha2, bias2, out, T);
}